// Summarizer_44195213476616
// MI455X (gfx1250) — compile-verified
//
#include <hip/hip_runtime.h>
#include <math.h>

// Problem dims
#define Bv   16
#define Sv   512
#define Tv   64
#define Ev   512
#define Hv   1024
#define Vv   32000

typedef float v2f __attribute__((ext_vector_type(2)));
typedef float v8f __attribute__((ext_vector_type(8)));

__device__ __forceinline__ v8f wmma_f32(v2f a, v2f b, v8f c) {
  // D = A(16x4) * B(4x16) + C(16x16), fp32, wave32
  return __builtin_amdgcn_wmma_f32_16x16x4_f32(false, a, false, b, (short)0, c,
                                               false, false);
}

__device__ __forceinline__ float sigm(float x) {
  return 1.0f / (1.0f + __expf(-x));
}

// ---------------------------------------------------------------------------
// Init: zero d_out[:,0,:], h0, c, barrier counters
// ---------------------------------------------------------------------------
__global__ void init_kernel(float* __restrict__ out, float* __restrict__ h0,
                            float* __restrict__ cb, unsigned* __restrict__ cnt) {
  int i = blockIdx.x * blockDim.x + threadIdx.x;
  int stride = gridDim.x * blockDim.x;
  for (int idx = i; idx < Bv * Vv; idx += stride) {
    int b = idx / Vv, v = idx % Vv;
    out[((size_t)b * Tv) * Vv + v] = 0.0f;  // t = 0 slab stays zero
  }
  for (int idx = i; idx < Bv * Hv; idx += stride) {
    h0[idx] = 0.0f;
    cb[idx] = 0.0f;
  }
  if (i < 4) cnt[i] = 0u;
}

// ---------------------------------------------------------------------------
// v1[j] = sum_h align_w[h] * attn_W[h][j],  j in [0,H)
// (the h-dependent half of attn_W and all bias terms are constant over s and
//  cancel in the softmax, so only this projection of attn_W matters)
// ---------------------------------------------------------------------------
__global__ void v1_kernel(const float* __restrict__ attn_W,
                          const float* __restrict__ align_w,
                          float* __restrict__ v1) {
  int j = blockIdx.x * blockDim.x + threadIdx.x;
  if (j >= Hv) return;
  float acc = 0.0f;
  for (int h = 0; h < Hv; ++h) acc += align_w[h] * attn_W[(size_t)h * (2 * Hv) + j];
  v1[j] = acc;
}

// ---------------------------------------------------------------------------
// Device-wide barrier: monotonically increasing arrive counter
// ---------------------------------------------------------------------------
__device__ __forceinline__ void grid_barrier(unsigned* cnt, unsigned target) {
  __syncthreads();
  if (threadIdx.x == 0) {
    __threadfence();  // release h writes
    atomicAdd(cnt, 1u);
    while (__hip_atomic_load(cnt, __ATOMIC_ACQUIRE, __HIP_MEMORY_SCOPE_AGENT) <
           target) {
      __builtin_amdgcn_s_sleep(1);
    }
  }
  __syncthreads();
  __threadfence();  // acquire for all threads
}

// ---------------------------------------------------------------------------
// One K-segment of the gate GEMM with 4 independent accumulator chains.
// ---------------------------------------------------------------------------
__device__ __forceinline__ void mma_seg(const float* __restrict__ arow,
                                        const float* __restrict__ brow,
                                        int klen, int koff, v8f& a0, v8f& a1,
                                        v8f& a2, v8f& a3) {
#pragma unroll 2
  for (int k = 0; k < klen; k += 16) {
    v2f x0 = *(const v2f*)(arow + k + koff);
    v2f w0 = *(const v2f*)(brow + k + koff);
    v2f x1 = *(const v2f*)(arow + k + 4 + koff);
    v2f w1 = *(const v2f*)(brow + k + 4 + koff);
    v2f x2 = *(const v2f*)(arow + k + 8 + koff);
    v2f w2 = *(const v2f*)(brow + k + 8 + koff);
    v2f x3 = *(const v2f*)(arow + k + 12 + koff);
    v2f w3 = *(const v2f*)(brow + k + 12 + koff);
    a0 = wmma_f32(x0, w0, a0);
    a1 = wmma_f32(x1, w1, a1);
    a2 = wmma_f32(x2, w2, a2);
    a3 = wmma_f32(x3, w3, a3);
  }
}

// ---------------------------------------------------------------------------
// Encoder: persistent kernel, 64 blocks x 128 threads (4 waves = 4 gates).
// Block handles h columns [blk*16, blk*16+16); wave g computes gate tile
// j = g*H + n_base. One WMMA tile covers the whole batch (M=16).
// h double-buffered across the grid barrier.
// ---------------------------------------------------------------------------
__global__ void __launch_bounds__(128) encoder_kernel(
    const int* __restrict__ source, const float* __restrict__ emb_enc,
    const float* __restrict__ Wih, const float* __restrict__ Whh,
    const float* __restrict__ bias, float* __restrict__ h0,
    float* __restrict__ h1, float* __restrict__ c_buf,
    float* __restrict__ enc_outs, unsigned* __restrict__ cnt) {
  __shared__ float g_lds[4][16][16];
  const int tid  = threadIdx.x;
  const int wave = tid >> 5;          // gate index 0..3 (i,f,g,o)
  const int lane = tid & 31;
  const int row  = lane & 15;         // M row (batch) / N col / weight row
  const int koff = (lane >> 4) << 1;  // 0 or 2 (K offset within fragment)
  const int mhi  = (lane >> 4) << 3;  // 0 or 8 (M offset of C/D fragment)
  const int n_base = blockIdx.x * 16;
  const int j = wave * Hv + n_base + row;  // gate weight row for B fragment
  const float* wih_row = Wih + (size_t)j * Ev;
  const float* whh_row = Whh + (size_t)j * Hv;
  const float  bval    = bias[j];
  const unsigned nblk  = gridDim.x;

  for (int t = 0; t < Sv; ++t) {
    const float* hr = (t & 1) ? h1 : h0;
    float*       hw = (t & 1) ? h0 : h1;
    const int    src  = source[row * Sv + t];
    const float* xrow = emb_enc + (size_t)src * Ev;
    const float* hrow = hr + row * Hv;

    v8f a0, a1, a2, a3;
#pragma unroll
    for (int r = 0; r < 8; ++r) {
      a0[r] = bval;
      a1[r] = 0.0f;
      a2[r] = 0.0f;
      a3[r] = 0.0f;
    }
    mma_seg(xrow, wih_row, Ev, koff, a0, a1, a2, a3);
    mma_seg(hrow, whh_row, Hv, koff, a0, a1, a2, a3);

#pragma unroll
    for (int r = 0; r < 8; ++r)
      g_lds[wave][r + mhi][row] = (a0[r] + a1[r]) + (a2[r] + a3[r]);
    __syncthreads();

    for (int idx = tid; idx < 256; idx += 128) {
      int bb = idx >> 4, n = idx & 15;
      float gi = g_lds[0][bb][n];
      float gf = g_lds[1][bb][n];
      float gg = g_lds[2][bb][n];
      float go = g_lds[3][bb][n];
      int off = bb * Hv + n_base + n;
      float cc = sigm(gf) * c_buf[off] + sigm(gi) * tanhf(gg);
      float hh = sigm(go) * tanhf(cc);
      c_buf[off] = cc;
      hw[off]    = hh;
      enc_outs[((size_t)t * Bv + bb) * Hv + n_base + n] = hh;
    }
    grid_barrier(cnt, (unsigned)(t + 1) * nblk);
  }
}

// ---------------------------------------------------------------------------
// Attention (step-invariant!): scores[b,s] = enc_outs[s,b,:].v1,
// softmax over s, ctx[b,:] = sum_s w[s]*enc_outs[s,b,:]. One block per b.
// ---------------------------------------------------------------------------
__global__ void __launch_bounds__(256) attn_kernel(
    const float* __restrict__ enc_outs, const float* __restrict__ v1,
    float* __restrict__ ctx) {
  __shared__ float sc[Sv];
  __shared__ float red[256];
  const int b = blockIdx.x;
  const int tid = threadIdx.x;

  for (int s = tid; s < Sv; s += 256) {
    const float* er = enc_outs + ((size_t)s * Bv + b) * Hv;
    float acc = 0.0f;
    for (int n = 0; n < Hv; ++n) acc += er[n] * v1[n];
    sc[s] = acc;
  }
  __syncthreads();

  // max
  red[tid] = fmaxf(sc[tid], sc[tid + 256]);
  __syncthreads();
  for (int off = 128; off; off >>= 1) {
    if (tid < off) red[tid] = fmaxf(red[tid], red[tid + off]);
    __syncthreads();
  }
  float mx = red[0];
  __syncthreads();

  // exp + sum
  float e0 = __expf(sc[tid] - mx);
  float e1 = __expf(sc[tid + 256] - mx);
  sc[tid]       = e0;
  sc[tid + 256] = e1;
  red[tid] = e0 + e1;
  __syncthreads();
  for (int off = 128; off; off >>= 1) {
    if (tid < off) red[tid] += red[tid + off];
    __syncthreads();
  }
  float inv = 1.0f / red[0];
  __syncthreads();
  sc[tid] *= inv;
  sc[tid + 256] *= inv;
  __syncthreads();

  for (int n = tid; n < Hv; n += 256) {
    float acc = 0.0f;
    for (int s = 0; s < Sv; ++s)
      acc += sc[s] * enc_outs[((size_t)s * Bv + b) * Hv + n];
    ctx[b * Hv + n] = acc;
  }
}

// ---------------------------------------------------------------------------
// Decoder: persistent, 63 steps. xt = [emb_dec[tok] (E) | ctx (H)] through
// dec_Wih (K=1536), plus h through dec_Whh (K=1024). Same gate layout.
// Writes all hidden states to hdec for the batched output GEMM.
// ---------------------------------------------------------------------------
__global__ void __launch_bounds__(128) decoder_kernel(
    const int* __restrict__ target, const float* __restrict__ emb_dec,
    const float* __restrict__ Wih, const float* __restrict__ Whh,
    const float* __restrict__ bias, const float* __restrict__ ctx,
    float* __restrict__ h0, float* __restrict__ h1, float* __restrict__ c_buf,
    float* __restrict__ hdec, unsigned* __restrict__ cnt) {
  __shared__ float g_lds[4][16][16];
  const int tid  = threadIdx.x;
  const int wave = tid >> 5;
  const int lane = tid & 31;
  const int row  = lane & 15;
  const int koff = (lane >> 4) << 1;
  const int mhi  = (lane >> 4) << 3;
  const int n_base = blockIdx.x * 16;
  const int j = wave * Hv + n_base + row;
  const float* wih_row = Wih + (size_t)j * (Ev + Hv);  // K = 1536
  const float* whh_row = Whh + (size_t)j * Hv;
  const float  bval    = bias[j];
  const float* crow    = ctx + row * Hv;
  const unsigned nblk  = gridDim.x;

  for (int t = 0; t < Tv - 1; ++t) {
    const float* hr = (t & 1) ? h1 : h0;  // t=0 reads encoder's final h (h0)
    float*       hw = (t & 1) ? h0 : h1;
    const int    tok  = target[row * Tv + t];
    const float* xrow = emb_dec + (size_t)tok * Ev;
    const float* hrow = hr + row * Hv;

    v8f a0, a1, a2, a3;
#pragma unroll
    for (int r = 0; r < 8; ++r) {
      a0[r] = bval;
      a1[r] = 0.0f;
      a2[r] = 0.0f;
      a3[r] = 0.0f;
    }
    mma_seg(xrow, wih_row, Ev, koff, a0, a1, a2, a3);       // embedding part
    mma_seg(crow, wih_row + Ev, Hv, koff, a0, a1, a2, a3);  // ctx part
    mma_seg(hrow, whh_row, Hv, koff, a0, a1, a2, a3);       // recurrent part

#pragma unroll
    for (int r = 0; r < 8; ++r)
      g_lds[wave][r + mhi][row] = (a0[r] + a1[r]) + (a2[r] + a3[r]);
    __syncthreads();

    for (int idx = tid; idx < 256; idx += 128) {
      int bb = idx >> 4, n = idx & 15;
      float gi = g_lds[0][bb][n];
      float gf = g_lds[1][bb][n];
      float gg = g_lds[2][bb][n];
      float go = g_lds[3][bb][n];
      int off = bb * Hv + n_base + n;
      float cc = sigm(gf) * c_buf[off] + sigm(gi) * tanhf(gg);
      float hh = sigm(go) * tanhf(cc);
      c_buf[off] = cc;
      hw[off]    = hh;
      hdec[((size_t)t * Bv + bb) * Hv + n_base + n] = hh;
    }
    grid_barrier(cnt, (unsigned)(t + 1) * nblk);
  }
}

// ---------------------------------------------------------------------------
// Output projection: preds[t][b][v] = hdec[t,b,:] . out_W[v,:] + out_b[v]
// Grid (250, 16): 8 waves/block, one 16-wide vocab tile per wave, 4 timesteps
// per block. Each B (weight) fragment is reused by 4 independent WMMA chains,
// cutting out_W L2 traffic 4x. Remainder t handled by clamping the A pointer
// (WMMA stays uniform; only stores are guarded).
// ---------------------------------------------------------------------------
__global__ void __launch_bounds__(256) out_gemm_kernel(
    const float* __restrict__ hdec, const float* __restrict__ out_W,
    const float* __restrict__ out_b, float* __restrict__ out) {
  const int tid  = threadIdx.x;
  const int wave = tid >> 5;
  const int lane = tid & 31;
  const int row  = lane & 15;
  const int koff = (lane >> 4) << 1;
  const int mhi  = (lane >> 4) << 3;
  const int t0    = blockIdx.y * 4;
  const int vbase = (blockIdx.x * 8 + wave) * 16;

  const float* wrow = out_W + (size_t)(vbase + row) * Hv;
  const float  bv   = out_b[vbase + row];

  const float* arow0 = hdec + ((size_t)min(t0 + 0, Tv - 2) * Bv + row) * Hv;
  const float* arow1 = hdec + ((size_t)min(t0 + 1, Tv - 2) * Bv + row) * Hv;
  const float* arow2 = hdec + ((size_t)min(t0 + 2, Tv - 2) * Bv + row) * Hv;
  const float* arow3 = hdec + ((size_t)min(t0 + 3, Tv - 2) * Bv + row) * Hv;

  v8f c0, c1, c2, c3;
#pragma unroll
  for (int r = 0; r < 8; ++r) {
    c0[r] = bv;
    c1[r] = bv;
    c2[r] = bv;
    c3[r] = bv;
  }

#pragma unroll 4
  for (int k = 0; k < Hv; k += 4) {
    v2f w = *(const v2f*)(wrow + k + koff);  // loaded once, used 4x
    v2f x0 = *(const v2f*)(arow0 + k + koff);
    v2f x1 = *(const v2f*)(arow1 + k + koff);
    v2f x2 = *(const v2f*)(arow2 + k + koff);
    v2f x3 = *(const v2f*)(arow3 + k + koff);
    c0 = wmma_f32(x0, w, c0);
    c1 = wmma_f32(x1, w, c1);
    c2 = wmma_f32(x2, w, c2);
    c3 = wmma_f32(x3, w, c3);
  }

#pragma unroll
  for (int r = 0; r < 8; ++r) {
    int m = r + mhi;  // batch index
    size_t col = (size_t)vbase + row;
    if (t0 + 0 < Tv - 1) out[((size_t)m * Tv + (t0 + 1)) * Vv + col] = c0[r];
    if (t0 + 1 < Tv - 1) out[((size_t)m * Tv + (t0 + 2)) * Vv + col] = c1[r];
    if (t0 + 2 < Tv - 1) out[((size_t)m * Tv + (t0 + 3)) * Vv + col] = c2[r];
    if (t0 + 3 < Tv - 1) out[((size_t)m * Tv + (t0 + 4)) * Vv + col] = c3[r];
  }
}

// ---------------------------------------------------------------------------
extern "C" void kernel_launch(void* const* d_in, const int* in_sizes, int n_in,
                              void* d_out, int out_size, void* d_ws,
                              size_t ws_size, hipStream_t stream) {
  const int*   source  = (const int*)d_in[0];
  const int*   target  = (const int*)d_in[1];
  const float* emb_enc = (const float*)d_in[2];
  const float* enc_Wih = (const float*)d_in[3];
  const float* enc_Whh = (const float*)d_in[4];
  const float* enc_b   = (const float*)d_in[5];
  const float* attn_W  = (const float*)d_in[6];
  // d_in[7] = attn_b, d_in[9] = align_b: constant over s -> cancel in softmax
  const float* align_w = (const float*)d_in[8];
  const float* emb_dec = (const float*)d_in[10];
  const float* dec_Wih = (const float*)d_in[11];
  const float* dec_Whh = (const float*)d_in[12];
  const float* dec_b   = (const float*)d_in[13];
  const float* out_W   = (const float*)d_in[14];
  const float* out_b   = (const float*)d_in[15];
  float* out = (float*)d_out;

  // Workspace layout (floats, after 256B counter region): ~36.2 MB total
  unsigned* cnt  = (unsigned*)d_ws;  // [0]=enc barrier, [1]=dec barrier
  float* base    = (float*)((char*)d_ws + 256);
  float* h0       = base;                 // 16K
  float* h1       = base + 16384;         // 16K
  float* cb       = base + 32768;         // 16K
  float* v1       = base + 49152;         // 1K
  float* ctx      = base + 50176;         // 16K
  float* hdec     = base + 66560;         // 63*16*1024 = 1,032,192
  float* enc_outs = base + 1098752;       // 512*16*1024 = 8,388,608

  init_kernel<<<dim3(512), dim3(256), 0, stream>>>(out, h0, cb, cnt);
  v1_kernel<<<dim3(4), dim3(256), 0, stream>>>(attn_W, align_w, v1);
  encoder_kernel<<<dim3(64), dim3(128), 0, stream>>>(
      source, emb_enc, enc_Wih, enc_Whh, enc_b, h0, h1, cb, enc_outs, cnt + 0);
  attn_kernel<<<dim3(16), dim3(256), 0, stream>>>(enc_outs, v1, ctx);
  decoder_kernel<<<dim3(64), dim3(128), 0, stream>>>(
      target, emb_dec, dec_Wih, dec_Whh, dec_b, ctx, h0, h1, cb, hdec, cnt + 1);
  out_gemm_kernel<<<dim3(250, 16), dim3(256), 0, stream>>>(hdec, out_W, out_b,
                                                           out);
}